// MTRRNet_68315749810369
// MI455X (gfx1250) — compile-verified
//
#include <hip/hip_runtime.h>
#include <hip/hip_bf16.h>
#include <math.h>

// ---------------- problem constants ----------------
#define BATCH   2
#define C_DIM   64
#define HW      16384          // 128*128
#define L_SEQ   16384
#define D_INNER 128
#define D_STATE 16
#define DT_RANK 4
#define XDBL_PAD 48            // 36 padded to 48
#define M_ROWS  (BATCH * L_SEQ)   // 32768
#define M_TILES (M_ROWS / 16)     // 2048
#define NCHUNK  64
#define CHUNK   (L_SEQ / NCHUNK)  // 256

typedef __attribute__((ext_vector_type(2))) float v2f;
typedef __attribute__((ext_vector_type(8))) float v8f;

// ---------------- FP32 WMMA 16x16 tile: D = A(16xK) * W(NxK)^T ----------------
// A layout  (ISA 7.12.2, 32-bit A 16x4):  lane -> M=lane&15, K = 2*(lane>>4)+{0,1}
// B layout  (4x16, row striped across lanes): vgpr v -> K = v + 2*(lane>>4), N = lane&15
// C/D layout: vgpr v -> M = v + 8*(lane>>4), N = lane&15
__device__ __forceinline__ v8f wmma_tile_f32(const float* __restrict__ act,
                                             const float* __restrict__ w,
                                             int mtile, int ntile, int K,
                                             int actStride, int wStride, int wRows) {
  const int lane = threadIdx.x & 31;
  const int r16  = lane & 15;
  const int half = lane >> 4;
  const float* arow = act + (size_t)(mtile * 16 + r16) * actStride;
  int wr = ntile * 16 + r16;
  if (wr >= wRows) wr = wRows - 1;   // clamp for padded N tiles (masked at store)
  const float* brow = w + (size_t)wr * wStride;
  v8f c = {0.f, 0.f, 0.f, 0.f, 0.f, 0.f, 0.f, 0.f};
  for (int k0 = 0; k0 < K; k0 += 4) {
    const float2 af = *(const float2*)(arow + k0 + 2 * half);
    const float2 bf = *(const float2*)(brow + k0 + 2 * half);
    v2f a; a.x = af.x; a.y = af.y;
    v2f b; b.x = bf.x; b.y = bf.y;
    c = __builtin_amdgcn_wmma_f32_16x16x4_f32(false, a, false, b, (short)0, c,
                                              false, false);
  }
  return c;
}

__device__ __forceinline__ float silu_f(float x)      { return x / (1.f + __expf(-x)); }
__device__ __forceinline__ float softplus_f(float x)  {
  return (x > 0.f) ? (x + log1pf(__expf(-x))) : log1pf(__expf(x));
}

// ---------------- 1) per-batch mean/var: deterministic 2-stage ----------------
__global__ void mtrr_stats_partial(const float* __restrict__ x, float* __restrict__ partial) {
  const int NPB = C_DIM * HW;                    // 1,048,576 per batch
  int b   = blockIdx.x >> 6;                     // 64 blocks / batch
  int blk = blockIdx.x & 63;
  float s = 0.f, ss = 0.f;
  for (int i = blk * 256 + threadIdx.x; i < NPB; i += 64 * 256) {
    float v = x[(size_t)b * NPB + i];
    s += v; ss += v * v;
  }
  for (int off = 16; off > 0; off >>= 1) {
    s  += __shfl_down(s,  off, 32);
    ss += __shfl_down(ss, off, 32);
  }
  __shared__ float sh[16];
  int lane = threadIdx.x & 31, w = threadIdx.x >> 5;
  if (lane == 0) { sh[w] = s; sh[w + 8] = ss; }
  __syncthreads();
  if (threadIdx.x == 0) {
    float ts = 0.f, tss = 0.f;
    for (int i = 0; i < 8; i++) { ts += sh[i]; tss += sh[i + 8]; }
    partial[blockIdx.x * 2 + 0] = ts;
    partial[blockIdx.x * 2 + 1] = tss;
  }
}

__global__ void mtrr_stats_finalize(const float* __restrict__ partial, float* __restrict__ stats) {
  if (threadIdx.x == 0) {
    for (int b = 0; b < BATCH; b++) {
      float s = 0.f, ss = 0.f;
      for (int i = 0; i < 64; i++) {
        s  += partial[(b * 64 + i) * 2 + 0];
        ss += partial[(b * 64 + i) * 2 + 1];
      }
      stats[b * 2 + 0] = s;
      stats[b * 2 + 1] = ss;
    }
  }
}

// ---------------- 2) normalize + clip ; emit residual (BCHW) and seq (BLC) ----------------
__global__ void mtrr_gn_normalize(const float* __restrict__ x, const float* __restrict__ gamma,
                                  const float* __restrict__ beta, const float* __restrict__ stats,
                                  float* __restrict__ res, float* __restrict__ seq) {
  int idx = blockIdx.x * 256 + threadIdx.x;      // 2*64*16384 = 2^21
  int b  = idx >> 20;
  int c  = (idx >> 14) & 63;
  int hw = idx & (HW - 1);
  const float invN = 1.f / (float)(C_DIM * HW);
  float mu  = stats[b * 2 + 0] * invN;
  float var = stats[b * 2 + 1] * invN - mu * mu;
  float rs  = rsqrtf(var + 1e-5f);
  float v = (x[idx] - mu) * rs * gamma[c] + beta[c];
  v = fminf(10.f, fmaxf(-10.f, v));
  res[idx] = v;
  seq[((size_t)(b * L_SEQ + hw)) * C_DIM + c] = v;
}

// ---------------- 3) in_proj GEMM: xz(BL,256) = seq(BL,64) x W(256,64)^T ----------------
__global__ void mtrr_gemm_in_proj(const float* __restrict__ seq, const float* __restrict__ w,
                                  float* __restrict__ xz) {
  int wave  = blockIdx.x * 8 + (threadIdx.x >> 5);
  int mtile = wave >> 4, ntile = wave & 15;      // 16 N-tiles
  v8f c = wmma_tile_f32(seq, w, mtile, ntile, 64, C_DIM, C_DIM, 256);
  int lane = threadIdx.x & 31, half = lane >> 4;
  int n = ntile * 16 + (lane & 15);
  #pragma unroll
  for (int v = 0; v < 8; v++) {
    int m = mtile * 16 + v + 8 * half;
    xz[(size_t)m * 256 + n] = c[v];
  }
}

// ---------------- 4) depthwise causal conv1d(k=4) + SiLU ----------------
__global__ void mtrr_conv_silu(const float* __restrict__ xz, const float* __restrict__ cw,
                               const float* __restrict__ cb, float* __restrict__ xm) {
  int idx = blockIdx.x * 256 + threadIdx.x;      // B*L*128 = 2^22
  int d  = idx & (D_INNER - 1);
  int bl = idx >> 7;
  int b  = bl >> 14;
  int l  = bl & (L_SEQ - 1);
  float acc = cb[d];
  #pragma unroll
  for (int j = 0; j < 4; j++) {
    int ls = l - 3 + j;
    if (ls >= 0)
      acc += xz[((size_t)(b * L_SEQ + ls)) * 256 + d] * cw[d * 4 + j];
  }
  xm[idx] = silu_f(acc);
}

// ---------------- 5) x_proj GEMM: xdbl(BL,48pad) = xm(BL,128) x W(36,128)^T ----------------
__global__ void mtrr_gemm_x_proj(const float* __restrict__ xm, const float* __restrict__ w,
                                 float* __restrict__ xdbl) {
  int wave  = blockIdx.x * 8 + (threadIdx.x >> 5);
  int mtile = wave / 3, ntile = wave % 3;        // 3 N-tiles cover 48 cols
  v8f c = wmma_tile_f32(xm, w, mtile, ntile, D_INNER, D_INNER, D_INNER, 36);
  int lane = threadIdx.x & 31, half = lane >> 4;
  int n = ntile * 16 + (lane & 15);
  if (n < 36) {
    #pragma unroll
    for (int v = 0; v < 8; v++) {
      int m = mtile * 16 + v + 8 * half;
      xdbl[(size_t)m * XDBL_PAD + n] = c[v];
    }
  }
}

// ---------------- 6) dt_proj GEMM (K=4) + softplus -> delta(BL,128) ----------------
__global__ void mtrr_gemm_dt(const float* __restrict__ xdbl, const float* __restrict__ w,
                             const float* __restrict__ dtb, float* __restrict__ delta) {
  int wave  = blockIdx.x * 8 + (threadIdx.x >> 5);
  int mtile = wave >> 3, ntile = wave & 7;       // 8 N-tiles
  v8f c = wmma_tile_f32(xdbl, w, mtile, ntile, DT_RANK, XDBL_PAD, DT_RANK, D_INNER);
  int lane = threadIdx.x & 31, half = lane >> 4;
  int n = ntile * 16 + (lane & 15);
  float bias = dtb[n];
  #pragma unroll
  for (int v = 0; v < 8; v++) {
    int m = mtile * 16 + v + 8 * half;
    delta[(size_t)m * D_INNER + n] = softplus_f(c[v] + bias);
  }
}

// ---------------- 7) chunked selective scan ----------------
// lane id bits: n[3:0] d[10:4] b[11] ch[17:12]
__global__ void mtrr_scan_passA(const float* __restrict__ delta, const float* __restrict__ xm,
                                const float* __restrict__ xdbl, const float* __restrict__ A_log,
                                float* __restrict__ Acum, float* __restrict__ Hend) {
  int id = blockIdx.x * 256 + threadIdx.x;       // 262144 lanes
  int n = id & 15, d = (id >> 4) & 127, b = (id >> 11) & 1, ch = id >> 12;
  float An = -__expf(A_log[d * D_STATE + n]);
  float h = 0.f, ac = 1.f;
  int base = b * L_SEQ + ch * CHUNK;
  for (int t = 0; t < CHUNK; t++) {
    int bl = base + t;
    float dv = delta[(size_t)bl * D_INNER + d];
    float u  = xm[(size_t)bl * D_INNER + d];
    float Bv = xdbl[(size_t)bl * XDBL_PAD + DT_RANK + n];
    float a = __expf(dv * An);
    h  = a * h + dv * Bv * u;
    ac = ac * a;
  }
  size_t sidx = ((size_t)((b * D_INNER + d) * D_STATE + n)) * NCHUNK + ch;
  Acum[sidx] = ac;
  Hend[sidx] = h;
}

__global__ void mtrr_scan_passB(const float* __restrict__ Acum, const float* __restrict__ Hend,
                                float* __restrict__ Hinit) {
  int id = blockIdx.x * 256 + threadIdx.x;       // 4096 lanes: (b,d,n)
  if (id >= BATCH * D_INNER * D_STATE) return;
  size_t base = (size_t)id * NCHUNK;
  float h = 0.f;
  for (int ch = 0; ch < NCHUNK; ch++) {
    Hinit[base + ch] = h;                        // exclusive prefix state
    h = Acum[base + ch] * h + Hend[base + ch];
  }
}

__global__ void mtrr_scan_passC(const float* __restrict__ delta, const float* __restrict__ xm,
                                const float* __restrict__ xdbl, const float* __restrict__ A_log,
                                const float* __restrict__ Dvec, const float* __restrict__ Hinit,
                                float* __restrict__ yscan) {
  int id = blockIdx.x * 256 + threadIdx.x;
  int n = id & 15, d = (id >> 4) & 127, b = (id >> 11) & 1, ch = id >> 12;
  float An = -__expf(A_log[d * D_STATE + n]);
  float Dd = Dvec[d];
  size_t sidx = ((size_t)((b * D_INNER + d) * D_STATE + n)) * NCHUNK + ch;
  float h = Hinit[sidx];
  int base = b * L_SEQ + ch * CHUNK;
  for (int t = 0; t < CHUNK; t++) {
    int bl = base + t;
    float dv = delta[(size_t)bl * D_INNER + d];
    float u  = xm[(size_t)bl * D_INNER + d];
    float Bv = xdbl[(size_t)bl * XDBL_PAD + DT_RANK + n];
    float Cv = xdbl[(size_t)bl * XDBL_PAD + DT_RANK + D_STATE + n];
    float a = __expf(dv * An);
    h = a * h + dv * Bv * u;
    float part = h * Cv;                          // reduce over 16 states
    part += __shfl_xor(part, 1, 32);
    part += __shfl_xor(part, 2, 32);
    part += __shfl_xor(part, 4, 32);
    part += __shfl_xor(part, 8, 32);
    if (n == 0) yscan[(size_t)bl * D_INNER + d] = part + u * Dd;
  }
}

// ---------------- 8) gate: ys = yscan * silu(z)  (overwrites xm) ----------------
__global__ void mtrr_gate(const float* __restrict__ yscan, const float* __restrict__ xz,
                          float* __restrict__ ys) {
  int idx = blockIdx.x * 256 + threadIdx.x;      // B*L*128
  int d  = idx & (D_INNER - 1);
  int bl = idx >> 7;
  float z = xz[(size_t)bl * 256 + 128 + d];
  ys[idx] = yscan[idx] * silu_f(z);
}

// ---------------- 9) out_proj GEMM: ym(BL,64) = ys(BL,128) x W(64,128)^T ----------------
__global__ void mtrr_gemm_out_proj(const float* __restrict__ ys, const float* __restrict__ w,
                                   float* __restrict__ ym) {
  int wave  = blockIdx.x * 8 + (threadIdx.x >> 5);
  int mtile = wave >> 2, ntile = wave & 3;       // 4 N-tiles
  v8f c = wmma_tile_f32(ys, w, mtile, ntile, D_INNER, D_INNER, D_INNER, C_DIM);
  int lane = threadIdx.x & 31, half = lane >> 4;
  int n = ntile * 16 + (lane & 15);
  #pragma unroll
  for (int v = 0; v < 8; v++) {
    int m = mtile * 16 + v + 8 * half;
    ym[(size_t)m * C_DIM + n] = c[v];
  }
}

// ---------------- 10) proj GEMM + BN + exact GELU + residual clip -> d_out ----------------
__global__ void mtrr_gemm_proj_fuse(const float* __restrict__ ym, const float* __restrict__ w,
                                    const float* __restrict__ pb, const float* __restrict__ bng,
                                    const float* __restrict__ bnb, const float* __restrict__ bnm,
                                    const float* __restrict__ bnv, const float* __restrict__ res,
                                    float* __restrict__ out) {
  int wave  = blockIdx.x * 8 + (threadIdx.x >> 5);
  int mtile = wave >> 2, ntile = wave & 3;       // 4 N-tiles
  v8f c = wmma_tile_f32(ym, w, mtile, ntile, C_DIM, C_DIM, C_DIM, C_DIM);
  int lane = threadIdx.x & 31, half = lane >> 4;
  int ch = ntile * 16 + (lane & 15);             // output channel o
  float bias = pb[ch];
  float rs   = rsqrtf(bnv[ch] + 1e-5f);
  float g    = bng[ch], be = bnb[ch], mn = bnm[ch];
  #pragma unroll
  for (int v = 0; v < 8; v++) {
    int m = mtile * 16 + v + 8 * half;
    int b = m >> 14;
    int l = m & (L_SEQ - 1);
    float p = c[v] + bias;
    p = (p - mn) * rs * g + be;
    float gl = 0.5f * p * (1.f + erff(p * 0.70710678118654752f));  // exact GELU
    size_t oidx = ((size_t)(b * C_DIM + ch)) * HW + l;
    float r = res[oidx] + gl;
    out[oidx] = fminf(10.f, fmaxf(-10.f, r));
  }
}

// ---------------- launch ----------------
extern "C" void kernel_launch(void* const* d_in, const int* in_sizes, int n_in,
                              void* d_out, int out_size, void* d_ws, size_t ws_size,
                              hipStream_t stream) {
  const float* x        = (const float*)d_in[0];
  const float* gn_gamma = (const float*)d_in[1];
  const float* gn_beta  = (const float*)d_in[2];
  const float* in_proj  = (const float*)d_in[3];
  const float* conv_w   = (const float*)d_in[4];
  const float* conv_b   = (const float*)d_in[5];
  const float* x_proj   = (const float*)d_in[6];
  const float* dt_w     = (const float*)d_in[7];
  const float* dt_b     = (const float*)d_in[8];
  const float* A_log    = (const float*)d_in[9];
  const float* Dvec     = (const float*)d_in[10];
  const float* out_w    = (const float*)d_in[11];
  const float* proj_w   = (const float*)d_in[12];
  const float* proj_b   = (const float*)d_in[13];
  const float* bn_g     = (const float*)d_in[14];
  const float* bn_b     = (const float*)d_in[15];
  const float* bn_m     = (const float*)d_in[16];
  const float* bn_v     = (const float*)d_in[17];
  float* out = (float*)d_out;

  char* p = (char*)d_ws;
  auto alloc = [&](size_t bytes) -> float* {
    float* r = (float*)p;
    p += (bytes + 255) & ~(size_t)255;
    return r;
  };
  float* partial = alloc(256 * sizeof(float));
  float* stats   = alloc(16 * sizeof(float));
  float* res     = alloc((size_t)BATCH * C_DIM * HW * 4);        // 8 MB
  float* seq     = alloc((size_t)M_ROWS * C_DIM * 4);            // 8 MB
  float* xz      = alloc((size_t)M_ROWS * 256 * 4);              // 32 MB
  float* xm      = alloc((size_t)M_ROWS * D_INNER * 4);          // 16 MB
  float* xdbl    = alloc((size_t)M_ROWS * XDBL_PAD * 4);         // 6 MB
  float* delta   = alloc((size_t)M_ROWS * D_INNER * 4);          // 16 MB
  float* yscan   = alloc((size_t)M_ROWS * D_INNER * 4);          // 16 MB
  float* ym      = alloc((size_t)M_ROWS * C_DIM * 4);            // 8 MB
  size_t scanN   = (size_t)BATCH * D_INNER * D_STATE * NCHUNK;   // 262144
  float* Acum    = alloc(scanN * 4);
  float* Hend    = alloc(scanN * 4);
  float* Hinit   = alloc(scanN * 4);

  mtrr_stats_partial <<<128, 256, 0, stream>>>(x, partial);
  mtrr_stats_finalize<<<1, 32, 0, stream>>>(partial, stats);
  mtrr_gn_normalize  <<<(BATCH * C_DIM * HW) / 256, 256, 0, stream>>>(
      x, gn_gamma, gn_beta, stats, res, seq);
  mtrr_gemm_in_proj  <<<(M_TILES * 16) / 8, 256, 0, stream>>>(seq, in_proj, xz);
  mtrr_conv_silu     <<<(M_ROWS * D_INNER) / 256, 256, 0, stream>>>(xz, conv_w, conv_b, xm);
  mtrr_gemm_x_proj   <<<(M_TILES * 3) / 8, 256, 0, stream>>>(xm, x_proj, xdbl);
  mtrr_gemm_dt       <<<(M_TILES * 8) / 8, 256, 0, stream>>>(xdbl, dt_w, dt_b, delta);
  mtrr_scan_passA    <<<(int)(scanN / 256), 256, 0, stream>>>(delta, xm, xdbl, A_log, Acum, Hend);
  mtrr_scan_passB    <<<16, 256, 0, stream>>>(Acum, Hend, Hinit);
  mtrr_scan_passC    <<<(int)(scanN / 256), 256, 0, stream>>>(delta, xm, xdbl, A_log, Dvec,
                                                              Hinit, yscan);
  mtrr_gate          <<<(M_ROWS * D_INNER) / 256, 256, 0, stream>>>(yscan, xz, xm);
  mtrr_gemm_out_proj <<<(M_TILES * 4) / 8, 256, 0, stream>>>(xm, out_w, ym);
  mtrr_gemm_proj_fuse<<<(M_TILES * 4) / 8, 256, 0, stream>>>(
      ym, proj_w, proj_b, bn_g, bn_b, bn_m, bn_v, res, out);
}